// ConjugateGradientSolver_90340342104330
// MI455X (gfx1250) — compile-verified
//
#include <hip/hip_runtime.h>
#include <cstdint>

// CG solve of (Laplacian(w)+I) x = b, COO SpMV with F=4 RHS columns.
// MI455X design notes:
//  - memory bound (0.23 FLOP/B); whole working set (~71MB) is L2-resident (192MB L2)
//  - SpMV: async-copy COO tiles to LDS (global_load_async_to_lds_b128, s_wait_asynccnt),
//    float4 gather of p, global_atomic_add_f32 scatter (resolves in L2)
//  - reductions: per-block partials (deterministic order), combined via
//    V_WMMA_F32_16X16X4_F32 with all-ones A (D = column-sums of B) on wave 0
//  - device-side `done` flag reproduces reference early termination under graph capture

#define CG_N 262144
#define CG_F 4
#define CG_NNZ 4262144      // 2*E + N
#define CG_RTOL 1e-5f
#define CG_MAXITER 100
#define TILE 2048
#define VEC_BLOCKS (CG_N / 256)   // 1024, one row (float4) per thread

typedef __attribute__((ext_vector_type(2))) float v2f;
typedef __attribute__((ext_vector_type(8))) float v8f;

// ---- WMMA sum of a 64-element chunk loaded as the B matrix ---------------
// A = ones(16x4)  =>  D[i][j] = sum_k B[k][j]  (all rows of D identical).
// sum_j D[0][j] over the 16 columns == sum of all 64 loaded elements,
// independent of the exact (k,j) <-> element bijection.

// Reduce 1024 partials (global memory) to one scalar, broadcast to the block.
// sdata needs >= 16 floats. Wave 0 does 16 chained WMMAs (accumulate via C).
__device__ inline float reduce_partials_1024(const float* __restrict__ part,
                                             float* sdata) {
  const int tid = threadIdx.x;
  __syncthreads();                 // protect any prior use of sdata
  if (tid < 32) {                  // wave 0 only: EXEC all-1s as WMMA requires
    const int n = tid & 15, g = tid >> 4;
    v8f c = {};
#pragma unroll
    for (int ch = 0; ch < 16; ++ch) {
      const int base = ch * 64;
      v2f a; a.x = 1.0f; a.y = 1.0f;
      v2f b;
      b.x = part[base + 32 * g + n];
      b.y = part[base + 32 * g + 16 + n];
      c = __builtin_amdgcn_wmma_f32_16x16x4_f32(false, a, false, b,
                                                (short)0, c, false, false);
    }
    sdata[n] = c[0];               // colsum_n (both half-waves write same value)
  }
  __syncthreads();
  float t = 0.0f;
#pragma unroll
  for (int j = 0; j < 16; ++j) t += sdata[j];
  return t;
}

// Block-wide sum of one float per thread (256 threads). sdata >= 272 floats.
__device__ inline float block_sum_256(float v, float* sdata) {
  const int tid = threadIdx.x;
  __syncthreads();                 // protect any prior use of sdata
  sdata[tid] = v;
  __syncthreads();
  if (tid < 32) {
    const int n = tid & 15, g = tid >> 4;
    v8f c = {};
#pragma unroll
    for (int ch = 0; ch < 4; ++ch) {
      const int base = ch * 64;
      v2f a; a.x = 1.0f; a.y = 1.0f;
      v2f b;
      b.x = sdata[base + 32 * g + n];
      b.y = sdata[base + 32 * g + 16 + n];
      c = __builtin_amdgcn_wmma_f32_16x16x4_f32(false, a, false, b,
                                                (short)0, c, false, false);
    }
    sdata[256 + n] = c[0];
  }
  __syncthreads();
  float t = 0.0f;
#pragma unroll
  for (int j = 0; j < 16; ++j) t += sdata[256 + j];
  return t;
}

// ---- kernels --------------------------------------------------------------

// x=0, r=b, p=b, q=0, part_a[bid] = block sum of b.b
__global__ void k_init(const float* __restrict__ b, float* __restrict__ x,
                       float* __restrict__ r, float* __restrict__ p,
                       float* __restrict__ q, float* __restrict__ part_a) {
  __shared__ float sdata[272];
  const int i = blockIdx.x * 256 + threadIdx.x;
  const float4 bv = reinterpret_cast<const float4*>(b)[i];
  float4 z; z.x = 0.f; z.y = 0.f; z.z = 0.f; z.w = 0.f;
  reinterpret_cast<float4*>(x)[i] = z;
  reinterpret_cast<float4*>(q)[i] = z;
  reinterpret_cast<float4*>(r)[i] = bv;
  reinterpret_cast<float4*>(p)[i] = bv;
  const float s = bv.x * bv.x + bv.y * bv.y + bv.z * bv.z + bv.w * bv.w;
  const float bs = block_sum_256(s, sdata);
  if (threadIdx.x == 0) part_a[blockIdx.x] = bs;
}

// rho = b.b ; tol2 = RTOL^2 * b.b ; done = (rho <= tol2)
__global__ void k_init2(const float* __restrict__ part_a,
                        float* __restrict__ scal, int* __restrict__ done) {
  __shared__ float sdata[272];
  const float bn2 = reduce_partials_1024(part_a, sdata);
  if (threadIdx.x == 0) {
    scal[0] = bn2;                      // rho
    scal[1] = CG_RTOL * CG_RTOL * bn2;  // tol^2
    *done = (bn2 <= scal[1]) ? 1 : 0;
  }
}

// loop head (iterations >= 1): rho = rho_new; convergence check
__global__ void k_head(const float* __restrict__ part_a,
                       float* __restrict__ scal, int* __restrict__ done) {
  if (*done) return;
  __shared__ float sdata[272];
  const float rn = reduce_partials_1024(part_a, sdata);
  if (threadIdx.x == 0) {
    scal[0] = rn;
    if (rn <= scal[1]) *done = 1;
  }
}

// q += A p : stage COO tile in LDS via async copy, gather p, atomic scatter
__global__ void k_spmv(const float* __restrict__ vals,
                       const int* __restrict__ row,
                       const int* __restrict__ col,
                       const float* __restrict__ p, float* __restrict__ q,
                       const int* __restrict__ done) {
  if (*done) return;
  __shared__ float s_val[TILE];
  __shared__ int s_col[TILE];
  __shared__ int s_row[TILE];

  const int tid = threadIdx.x;
  const int base = blockIdx.x * TILE;
  const int cnt = (CG_NNZ - base < TILE) ? (CG_NNZ - base) : TILE;

  if (cnt == TILE) {
    // CDNA5 async copy: 2 x b128 per thread per array (2048 * 4B = 8KB each)
    const uint32_t lv = (uint32_t)(uintptr_t)(void*)&s_val[0];
    const uint32_t lc = (uint32_t)(uintptr_t)(void*)&s_col[0];
    const uint32_t lr = (uint32_t)(uintptr_t)(void*)&s_row[0];
    const uint64_t gv = (uint64_t)(uintptr_t)(const void*)(vals + base);
    const uint64_t gc = (uint64_t)(uintptr_t)(const void*)(col + base);
    const uint64_t gr = (uint64_t)(uintptr_t)(const void*)(row + base);
#pragma unroll
    for (int h = 0; h < 2; ++h) {
      const uint32_t boff = (uint32_t)(tid + h * 256) * 16u;
      asm volatile("global_load_async_to_lds_b128 %0, %1, off"
                   :: "v"(lv + boff), "v"(gv + boff) : "memory");
      asm volatile("global_load_async_to_lds_b128 %0, %1, off"
                   :: "v"(lc + boff), "v"(gc + boff) : "memory");
      asm volatile("global_load_async_to_lds_b128 %0, %1, off"
                   :: "v"(lr + boff), "v"(gr + boff) : "memory");
    }
    asm volatile("s_wait_asynccnt 0x0" ::: "memory");
  } else {
    for (int idx = tid; idx < cnt; idx += 256) {
      s_val[idx] = vals[base + idx];
      s_col[idx] = col[base + idx];
      s_row[idx] = row[base + idx];
    }
  }
  __syncthreads();

  const float4* p4 = reinterpret_cast<const float4*>(p);
#pragma unroll
  for (int h = 0; h < 8; ++h) {
    const int idx = tid + h * 256;
    if (idx < cnt) {
      const float v = s_val[idx];
      const int c = s_col[idx];
      const int rr = s_row[idx];
      const float4 pc = p4[c];
      float* qr = q + 4 * rr;
      unsafeAtomicAdd(qr + 0, v * pc.x);
      unsafeAtomicAdd(qr + 1, v * pc.y);
      unsafeAtomicAdd(qr + 2, v * pc.z);
      unsafeAtomicAdd(qr + 3, v * pc.w);
    }
  }
}

// part_b[bid] = block sum of p.q
__global__ void k_dot(const float* __restrict__ p, const float* __restrict__ q,
                      float* __restrict__ part_b,
                      const int* __restrict__ done) {
  if (*done) return;
  __shared__ float sdata[272];
  const int i = blockIdx.x * 256 + threadIdx.x;
  const float4 pi = reinterpret_cast<const float4*>(p)[i];
  const float4 qi = reinterpret_cast<const float4*>(q)[i];
  const float s = pi.x * qi.x + pi.y * qi.y + pi.z * qi.z + pi.w * qi.w;
  const float bs = block_sum_256(s, sdata);
  if (threadIdx.x == 0) part_b[blockIdx.x] = bs;
}

// alpha = rho / (p.q); x += alpha p; r -= alpha q; q = 0; part_a[bid] = blk(r.r)
__global__ void k_update(float* __restrict__ x, float* __restrict__ r,
                         const float* __restrict__ p, float* __restrict__ q,
                         const float* __restrict__ part_b,
                         const float* __restrict__ scal,
                         float* __restrict__ part_a,
                         const int* __restrict__ done) {
  if (*done) return;
  __shared__ float sdata[272];
  const float pq = reduce_partials_1024(part_b, sdata);
  const float alpha = scal[0] / pq;
  const int i = blockIdx.x * 256 + threadIdx.x;
  const float4 pi = reinterpret_cast<const float4*>(p)[i];
  const float4 qi = reinterpret_cast<const float4*>(q)[i];
  float4 xi = reinterpret_cast<float4*>(x)[i];
  float4 ri = reinterpret_cast<float4*>(r)[i];
  xi.x += alpha * pi.x; xi.y += alpha * pi.y;
  xi.z += alpha * pi.z; xi.w += alpha * pi.w;
  ri.x -= alpha * qi.x; ri.y -= alpha * qi.y;
  ri.z -= alpha * qi.z; ri.w -= alpha * qi.w;
  reinterpret_cast<float4*>(x)[i] = xi;
  reinterpret_cast<float4*>(r)[i] = ri;
  float4 z; z.x = 0.f; z.y = 0.f; z.z = 0.f; z.w = 0.f;
  reinterpret_cast<float4*>(q)[i] = z;   // pre-zero q for the next SpMV
  const float rs = ri.x * ri.x + ri.y * ri.y + ri.z * ri.z + ri.w * ri.w;
  const float bs = block_sum_256(rs, sdata);
  if (threadIdx.x == 0) part_a[blockIdx.x] = bs;
}

// beta = rho_new / rho; p = r + beta p
__global__ void k_pupdate(const float* __restrict__ r, float* __restrict__ p,
                          const float* __restrict__ part_a,
                          const float* __restrict__ scal,
                          const int* __restrict__ done) {
  if (*done) return;
  __shared__ float sdata[272];
  const float rn = reduce_partials_1024(part_a, sdata);
  const float beta = rn / scal[0];
  const int i = blockIdx.x * 256 + threadIdx.x;
  const float4 ri = reinterpret_cast<const float4*>(r)[i];
  float4 pi = reinterpret_cast<float4*>(p)[i];
  pi.x = ri.x + beta * pi.x; pi.y = ri.y + beta * pi.y;
  pi.z = ri.z + beta * pi.z; pi.w = ri.w + beta * pi.w;
  reinterpret_cast<float4*>(p)[i] = pi;
}

// ---- host -----------------------------------------------------------------

extern "C" void kernel_launch(void* const* d_in, const int* in_sizes, int n_in,
                              void* d_out, int out_size, void* d_ws,
                              size_t ws_size, hipStream_t stream) {
  const float* vals = (const float*)d_in[0];   // [NNZ]
  const float* b = (const float*)d_in[1];      // [N, 4]
  const int* row = (const int*)d_in[2];        // [NNZ]
  const int* col = (const int*)d_in[3];        // [NNZ]
  float* x = (float*)d_out;                    // [N, 4]

  const int NF = CG_N * CG_F;
  float* wsf = (float*)d_ws;
  float* q = wsf;
  float* r = wsf + NF;
  float* p = wsf + 2 * NF;
  float* part_a = wsf + 3 * NF;        // [1024]
  float* part_b = part_a + 1024;       // [1024]
  float* scal = part_b + 1024;         // [0]=rho, [1]=tol^2
  int* done = (int*)(scal + 8);

  const int spmv_blocks = (CG_NNZ + TILE - 1) / TILE;

  k_init<<<VEC_BLOCKS, 256, 0, stream>>>(b, x, r, p, q, part_a);
  k_init2<<<1, 256, 0, stream>>>(part_a, scal, done);

  for (int t = 0; t < CG_MAXITER; ++t) {
    if (t > 0) k_head<<<1, 256, 0, stream>>>(part_a, scal, done);
    k_spmv<<<spmv_blocks, 256, 0, stream>>>(vals, row, col, p, q, done);
    k_dot<<<VEC_BLOCKS, 256, 0, stream>>>(p, q, part_b, done);
    k_update<<<VEC_BLOCKS, 256, 0, stream>>>(x, r, p, q, part_b, scal, part_a,
                                             done);
    k_pupdate<<<VEC_BLOCKS, 256, 0, stream>>>(r, p, part_a, scal, done);
  }
}